// HashGrid2D_53120155517403
// MI455X (gfx1250) — compile-verified
//
#include <hip/hip_runtime.h>

// ---------------------------------------------------------------------------
// HashGrid2D encode (Instant-NGP style), MI455X / gfx1250.
//
// Memory-bound gather kernel (no matmul -> WMMA not applicable):
//   - tables (64 MB) kept resident in the 192 MB L2: table gathers use default
//     (RT) temporal policy; ~1 GB of 8B gathers run at L2 bandwidth.
//   - output (256 MB write-once stream > L2 size) uses NON-TEMPORAL stores so
//     it cannot evict the table working set.
//   - xy (16 MB read-once stream) is staged through LDS with a wave-local
//     ASYNC double-buffer: global_load_async_to_lds_b64 for tile i+1 is issued
//     while tile i computes; s_wait_asynccnt 0x1 retires tile i's stage
//     (ASYNC ops complete in order). Barrier-free: each wave owns its slots.
//   - global_prefetch_b8 warms L2 two tiles ahead of the async stage.
//
// Lane mapping (wave32): lane&15 = level, lane>>4 = point-within-pair.
//   -> one wave = 2 points x 16 levels; the 8-byte/lane output store is a
//      fully coalesced 256B contiguous write per wave.
// ---------------------------------------------------------------------------

typedef float v2f __attribute__((ext_vector_type(2)));

#define NLEVELS   16
#define TBL_SIZE  (1u << 19)
#define TBL_MASK  ((1u << 19) - 1u)
#define PRIME_Y   2654435761u

// round(16 * (128^(1/15))^l), l = 0..15  (matches the Python reference)
__constant__ int c_res[NLEVELS] = {
    16, 22, 31, 42, 58, 81, 111, 154, 213, 294, 406, 562, 776, 1072, 1482, 2048
};

// Generic (__shared__) pointer -> 32-bit LDS byte offset for async-LDS VDST.
#define LDS_OFFSET(p) \
    ((unsigned)(unsigned long long)(__attribute__((address_space(3))) const void*)(p))

__global__ __launch_bounds__(256)
void hashgrid2d_encode(const float* __restrict__ xy,
                       const float* __restrict__ tables,
                       float* __restrict__ out,
                       int n_points, int n_tiles) {
#pragma clang fp contract(off)   // match reference mul/mul/add rounding exactly
    __shared__ v2f stage[2][8][2];   // [buf][wave][point-in-pair], 256 B total

    const int lane       = threadIdx.x & 31;
    const int wave       = threadIdx.x >> 5;           // 0..7
    const int level      = threadIdx.x & 15;           // 0..15
    const int pt_sel     = lane >> 4;                  // 0/1 within the wave's pair
    const int pt_in_tile = threadIdx.x >> 4;           // 0..15 (16 points/tile)
    const int Ni         = c_res[level];
    const float Nf       = (float)Ni;
    const float* __restrict__ tab = tables + (size_t)level * (TBL_SIZE * 2u);

    const int stride = (int)gridDim.x;
    int tile = (int)blockIdx.x;
    if (tile >= n_tiles) return;

    // Prologue: async-stage this wave's two xy points for tile0 into buf 0.
    if (lane < 2) {
        const unsigned dst = LDS_OFFSET(&stage[0][wave][lane]);
        const unsigned long long src = (unsigned long long)(const void*)
            (xy + ((size_t)tile * 16u + (size_t)wave * 2u + (unsigned)lane) * 2u);
        asm volatile("global_load_async_to_lds_b64 %0, %1, off"
                     :: "v"(dst), "v"(src) : "memory");
    }

    int buf = 0;
    for (; tile < n_tiles; tile += stride) {
        // Kick off the async stage for the NEXT tile into the other buffer
        // (index clamped so the tail iteration never reads out of bounds).
        const int next   = tile + stride;
        const int next_c = (next < n_tiles) ? next : tile;
        if (lane < 2) {
            const unsigned dst = LDS_OFFSET(&stage[buf ^ 1][wave][lane]);
            const unsigned long long src = (unsigned long long)(const void*)
                (xy + ((size_t)next_c * 16u + (size_t)wave * 2u + (unsigned)lane) * 2u);
            asm volatile("global_load_async_to_lds_b64 %0, %1, off"
                         :: "v"(dst), "v"(src) : "memory");
        }
        // Warm L2 two tiles ahead so the async stage itself is a cache hit.
        if (threadIdx.x == 0) {
            const int pf = tile + 2 * stride;
            if (pf < n_tiles) __builtin_prefetch(xy + (size_t)pf * 32u, 0, 0);
        }
        // <=1 outstanding async op  =>  current tile's stage (issued last
        // iteration / prologue) has landed in LDS. In-order completion.
        asm volatile("s_wait_asynccnt 0x1" ::: "memory");

        const int point = tile * 16 + pt_in_tile;
        if (point < n_points) {
            const v2f p = stage[buf][wave][pt_sel];   // conflict-free ds broadcast

            const float sx = p.x * Nf;
            const float sy = p.y * Nf;
            const float fx = floorf(sx);
            const float fy = floorf(sy);
            const float wx = sx - fx;
            const float wy = sy - fy;
            int ix = (int)fx;                 // in [0, Ni] (== Ni only via rounding)
            int iy = (int)fy;

            // (i + d) % Ni with i in [0, Ni]: one compare/sub, no divide.
            int ix0 = (ix >= Ni) ? ix - Ni : ix;
            int iy0 = (iy >= Ni) ? iy - Ni : iy;
            int ix1 = ix0 + 1; ix1 = (ix1 >= Ni) ? ix1 - Ni : ix1;
            int iy1 = iy0 + 1; iy1 = (iy1 >= Ni) ? iy1 - Ni : iy1;

            // hash = (cx * 1) ^ (cy * 2654435761)  mod 2^19  (mask, no divide)
            const unsigned hy0 = (unsigned)iy0 * PRIME_Y;
            const unsigned hy1 = (unsigned)iy1 * PRIME_Y;
            const unsigned h00 = ((unsigned)ix0 ^ hy0) & TBL_MASK;
            const unsigned h10 = ((unsigned)ix1 ^ hy0) & TBL_MASK;
            const unsigned h01 = ((unsigned)ix0 ^ hy1) & TBL_MASK;
            const unsigned h11 = ((unsigned)ix1 ^ hy1) & TBL_MASK;

            // 4 independent 8B gathers per lane; L2-resident tables.
            const v2f c00 = *(const v2f*)(tab + 2u * h00);
            const v2f c10 = *(const v2f*)(tab + 2u * h10);
            const v2f c01 = *(const v2f*)(tab + 2u * h01);
            const v2f c11 = *(const v2f*)(tab + 2u * h11);

            const float owx = 1.0f - wx;
            const float owy = 1.0f - wy;
            const v2f f0 = c00 * owx + c10 * wx;   // exact reference expr order
            const v2f f1 = c01 * owx + c11 * wx;
            const v2f r  = f0 * owy + f1 * wy;

            // Non-temporal: 256MB stream must not thrash L2.
            // out[point*32 + level*2 .. +1]  == v2f index point*16+level.
            __builtin_nontemporal_store(r, (v2f*)out + ((size_t)point * 16u + level));
        }
        buf ^= 1;
    }
}

extern "C" void kernel_launch(void* const* d_in, const int* in_sizes, int n_in,
                              void* d_out, int out_size, void* d_ws, size_t ws_size,
                              hipStream_t stream) {
    const float* xy     = (const float*)d_in[0];   // [N_POINTS, 2] f32
    const float* tables = (const float*)d_in[1];   // [16, 2^19, 2] f32
    float* out          = (float*)d_out;           // [N_POINTS, 32] f32

    const int n_points = in_sizes[0] / 2;
    const int n_tiles  = (n_points + 15) / 16;     // 16 points per 256-thread tile

    int grid = n_tiles < 8192 ? n_tiles : 8192;    // grid-stride; ~64K waves total
    if (grid < 1) grid = 1;

    hipLaunchKernelGGL(hashgrid2d_encode, dim3(grid), dim3(256), 0, stream,
                       xy, tables, out, n_points, n_tiles);
}